// FastfoodProjection_14534169329869
// MI455X (gfx1250) — compile-verified
//
#include <hip/hip_runtime.h>

#define NPAD (1 << 22)
#define DIN  1024
#define NOUT 4000000
#define NB   8

typedef __attribute__((ext_vector_type(2))) float v2f;
typedef __attribute__((ext_vector_type(8))) float v8f;

// Low 32 bits of a generic pointer into LDS == byte offset from wave LDS base
// (LDS aperture truncation, ISA 10.2).
__device__ __forceinline__ unsigned lds_u32(const void* p) {
  return (unsigned)(unsigned long long)p;
}

// ---------------------------------------------------------------------------
// Per-lane A-fragments of H16 for V_WMMA_F32_16X16X4_F32.
// A layout (16x4 f32): lane L -> M = L&15 ; VGPR v -> K = v + 2*(L>>4).
// H16[m][k] = (-1)^popcount(m&k), built as 0x3f800000 | parity<<31 (no cndmask).
// ---------------------------------------------------------------------------
__device__ __forceinline__ void make_hA(int lane, v2f* hA) {
  const int m = lane & 15, hh = lane >> 4;
#pragma unroll
  for (int kk = 0; kk < 4; ++kk) {
    const int k0 = 4 * kk + 2 * hh;
    hA[kk].x = __int_as_float(0x3f800000 |
                              ((__builtin_popcount(m & k0) & 1) << 31));
    hA[kk].y = __int_as_float(0x3f800000 |
                              ((__builtin_popcount(m & (k0 + 1)) & 1) << 31));
  }
}

// ---------------------------------------------------------------------------
// D = H16 * X, X read from LDS with row stride rs (floats).
// B layout (4x16 f32): lane L -> N = L&15, VGPR v -> K = v + 2*(L>>4).
// Four chained 16x16x4 WMMAs accumulate K=0..15.
// D/C layout: lane L holds (M = v + 8*(L>>4), N = L&15) in VGPR v.
// ---------------------------------------------------------------------------
__device__ __forceinline__ v8f hmulL(const float* src, int rs, int lane,
                                     const v2f* hA) {
  const int hh = lane >> 4, n = lane & 15;
  v8f acc = {};
#pragma unroll
  for (int kk = 0; kk < 4; ++kk) {
    v2f bf;
    bf.x = src[(4 * kk + 2 * hh + 0) * rs + n];
    bf.y = src[(4 * kk + 2 * hh + 1) * rs + n];
    acc = __builtin_amdgcn_wmma_f32_16x16x4_f32(
        false, hA[kk], false, bf, (short)0, acc, false, false);
  }
  return acc;
}

// ---------------------------------------------------------------------------
// In-place FWHT over 2048 contiguous floats in LDS, executed by ONE wave32.
// Stages 0..7: per 256-block Y = H16*X*H16 via WMMA (transpose through a
// 17-padded scratch -> bank-conflict-free).  Stages 8..10: radix-2
// butterflies, wave-private (LDS in-order within a wave, no barrier).
// ---------------------------------------------------------------------------
__device__ __forceinline__ void wave_fwht2048(float* t, float* scr, int lane,
                                              const v2f* hA) {
  const int hh = lane >> 4, n = lane & 15;
#pragma unroll
  for (int blk = 0; blk < 8; ++blk) {
    float* xb = t + blk * 256;
    v8f T = hmulL(xb, 16, lane, hA);           // T = H * X
#pragma unroll
    for (int v = 0; v < 8; ++v)                // scr[a*17+b] = T[b][a]
      scr[n * 17 + v + 8 * hh] = T[v];
    v8f U = hmulL(scr, 17, lane, hA);          // U = H * T^T
#pragma unroll
    for (int v = 0; v < 8; ++v)                // Y[n][m] = U[m][n]
      xb[n * 16 + v + 8 * hh] = U[v];
  }
  for (int s = 8; s <= 10; ++s) {
    const int st = 1 << s;
    for (int q = lane; q < 1024; q += 32) {
      const int idx = ((q >> s) << (s + 1)) | (q & (st - 1));
      const float a = t[idx], b = t[idx + st];
      t[idx] = a + b;
      t[idx + st] = a - b;
    }
  }
}

// ---------------------------------------------------------------------------
// Kernel 1: z[b] = FWHT_1024(x[b] * B[:1024]).  One WG per batch row.
// ---------------------------------------------------------------------------
__global__ __launch_bounds__(512) void k_fwht1024(const float* __restrict__ x,
                                                  const float* __restrict__ Bs,
                                                  float* __restrict__ z) {
  __shared__ float l[1024];
  const int b = blockIdx.x, t = threadIdx.x;
  l[t]       = x[b * DIN + t]       * Bs[t];
  l[t + 512] = x[b * DIN + t + 512] * Bs[t + 512];
  __syncthreads();
  for (int s = 0; s < 10; ++s) {
    const int h = 1 << s;
    const int idx = ((t >> s) << (s + 1)) | (t & (h - 1));
    const float a = l[idx], c = l[idx + h];
    l[idx] = a + c;
    l[idx + h] = a - c;
    __syncthreads();
  }
  z[b * DIN + t]       = l[t];
  z[b * DIN + t + 512] = l[t + 512];
}

// ---------------------------------------------------------------------------
// Pass A: fused gather v[i] = z[Pi[i]&1023] * G[i] + FWHT stages 0..10
// (contiguous 2048-segments).  WG = 4 waves, one segment per wave.
// Result segment leaves via GLOBAL_STORE_ASYNC_FROM_LDS_B32 (LDS -> HBM
// direct, ASYNCcnt-tracked, no VGPR round trip).
// ---------------------------------------------------------------------------
__global__ __launch_bounds__(128) void k_passA(const float* __restrict__ z,
                                               const int* __restrict__ Pi,
                                               const float* __restrict__ G,
                                               float* __restrict__ buf) {
  __shared__ float tile[4 * 2048];
  __shared__ float zl[1024];
  __shared__ float scr[4 * 272];
  const int b = blockIdx.y, t = threadIdx.x;
  const int lane = t & 31, w = t >> 5;
  const int i0 = blockIdx.x * 8192;

  for (int j = t; j < 1024; j += 128) zl[j] = z[b * DIN + j];
  __syncthreads();
  for (int j = t; j < 8192; j += 128) {        // coalesced Pi/G stream
    const int i = i0 + j;
    tile[j] = zl[Pi[i] & 1023] * G[i];
  }
  __syncthreads();

  v2f hA[4];
  make_hA(lane, hA);
  wave_fwht2048(tile + w * 2048, scr + w * 272, lane, hA);

  // LDS -> global async store of this wave's segment (64 x B32 per lane set).
  const float* sbase = buf + (b * NPAD + i0);  // uniform -> SGPR pair
  asm volatile("s_wait_dscnt 0x0" ::: "memory");  // DS writes before async read
  for (int k = lane; k < 2048; k += 32) {
    const unsigned voff = (unsigned)((w * 2048 + k) * 4);   // global byte off
    const unsigned laddr = lds_u32(&tile[w * 2048 + k]);    // LDS byte addr
    asm volatile("global_store_async_from_lds_b32 %0, %1, %2"
                 :: "v"(voff), "v"(laddr), "s"(sbase) : "memory");
  }
  asm volatile("s_wait_asynccnt 0x0" ::: "memory");
}

// ---------------------------------------------------------------------------
// Pass B: FWHT stages 11..21 (stride-2048 dimension).  The 2048x16 column
// tile is transpose-loaded straight into padded LDS (stride 2052 floats,
// conflict-free) with GLOBAL_LOAD_ASYNC_TO_LDS_B32 (per-lane scattered LDS
// destination, ASYNCcnt-tracked).  Each wave FWHTs 2 contiguous columns,
// scales by S, writes idx < 4M.  Dynamic LDS: 140032 B (< 320 KB/WG).
// ---------------------------------------------------------------------------
__global__ __launch_bounds__(256) void k_passB(const float* __restrict__ buf,
                                               const float* __restrict__ S,
                                               float* __restrict__ out) {
  extern __shared__ float sm[];
  const int CSTR = 2052;
  const int b = blockIdx.y, t = threadIdx.x;
  const int lane = t & 31, w = t >> 5;
  const int c0 = blockIdx.x * 16;
  const int cc = t & 15, rr0 = t >> 4;
  float* scr = sm + 16 * CSTR + w * 272;

  // global (row-major) -> LDS (column-major, padded) async transpose-load
  const float* sbase = buf + b * NPAD;         // uniform -> SGPR pair
  for (int r = rr0; r < 2048; r += 16) {
    const unsigned voff = (unsigned)((r * 2048 + c0 + cc) * 4);
    const unsigned laddr = lds_u32(&sm[cc * CSTR + r]);
    asm volatile("global_load_async_to_lds_b32 %0, %1, %2"
                 :: "v"(laddr), "v"(voff), "s"(sbase) : "memory");
  }
  asm volatile("s_wait_asynccnt 0x0" ::: "memory");
  __syncthreads();

  v2f hA[4];
  make_hA(lane, hA);
  wave_fwht2048(sm + (2 * w + 0) * CSTR, scr, lane, hA);
  wave_fwht2048(sm + (2 * w + 1) * CSTR, scr, lane, hA);
  __syncthreads();

  float* dst = out + b * NOUT;
  for (int r = rr0; r < 2048; r += 16) {
    const int idx = r * 2048 + c0 + cc;
    if (idx < NOUT) dst[idx] = sm[cc * CSTR + r] * S[idx];
  }
}

// ---------------------------------------------------------------------------
extern "C" void kernel_launch(void* const* d_in, const int* in_sizes, int n_in,
                              void* d_out, int out_size, void* d_ws,
                              size_t ws_size, hipStream_t stream) {
  const float* x  = (const float*)d_in[0];
  const float* Bs = (const float*)d_in[1];
  const float* G  = (const float*)d_in[2];
  const float* S  = (const float*)d_in[3];
  const int*   Pi = (const int*)d_in[4];
  float* out = (float*)d_out;

  float* z   = (float*)d_ws;            // 8 * 1024 floats
  float* buf = (float*)d_ws + 16384;    // 8 * 2^22 floats (128 MB)

  k_fwht1024<<<dim3(NB), dim3(512), 0, stream>>>(x, Bs, z);
  k_passA<<<dim3(512, NB), dim3(128), 0, stream>>>(z, Pi, G, buf);
  const size_t shB = (size_t)(16 * 2052 + 8 * 272) * sizeof(float);
  k_passB<<<dim3(128, NB), dim3(256), shB, stream>>>(buf, S, out);
}